// TAMoE_84997402788510
// MI455X (gfx1250) — compile-verified
//
#include <hip/hip_runtime.h>
#include <hip/hip_bf16.h>
#include <math.h>

// ---------------- problem constants ----------------
#define B_   32
#define L_   2048
#define D_   512
#define F_   1024
#define E_   8
#define DT_  16
#define NT_  (B_ * L_)      // 65536 tokens
#define MW_  32             // tokens per workgroup (2 WMMA M-tiles)
#define FC_  256            // F-chunk for GEMM1->GEMM2 pipelining through LDS
#define KC_  64             // K-chunk for async x staging (double buffered)
#define XSC_STRIDE 68       // 64 + 4 pad : stride%64==4 -> conflict-free column reads
#define HS_STRIDE  260      // 256 + 4 pad

typedef float v2f __attribute__((ext_vector_type(2)));
typedef float v8f __attribute__((ext_vector_type(8)));

// D = A(16x4 f32) * B(4x16 f32) + C(16x16 f32), full fp32 WMMA (CDNA5)
__device__ __forceinline__ v8f wmma4(v2f a, v2f b, v8f c) {
  return __builtin_amdgcn_wmma_f32_16x16x4_f32(
      /*neg_a=*/false, a, /*neg_b=*/false, b,
      /*c_mod=*/(short)0, c, /*reuse_a=*/false, /*reuse_b=*/false);
}

// async DMA one 16B chunk global -> LDS (no VGPR round trip, ASYNCcnt-tracked)
__device__ __forceinline__ void async_g2l_b128(void* lds_dst, const float* src) {
  const unsigned ldsoff = (unsigned)(uintptr_t)lds_dst;  // low 32 bits = LDS byte addr
  asm volatile("global_load_async_to_lds_b128 %0, %1, off"
               :: "v"(ldsoff), "v"(src) : "memory");
}
__device__ __forceinline__ void wait_async0() {
  asm volatile("s_wait_asynccnt 0x0" ::: "memory");
}

// ---------------- kernel 0: zero aux accumulators in ws ----------------
__global__ void tamoe_init(float* stats) {
  if (threadIdx.x < 16) stats[threadIdx.x] = 0.0f;
}

// ---------------- kernel 1: router statistics for aux loss ----------------
// stats[0..7]  = sum over tokens of softmax(logits)      (importance * NT)
// stats[8..15] = sum over tokens of top2 one-hot counts  (load * NT)
__global__ __launch_bounds__(256) void tamoe_router_stats(
    const float* __restrict__ x, const float* __restrict__ z_task,
    const float* __restrict__ tau, const float* __restrict__ Wr,
    const float* __restrict__ br, float* __restrict__ stats) {
  __shared__ float red[16];
  const int tid = threadIdx.x;
  if (tid < 16) red[tid] = 0.0f;
  __syncthreads();

  const int token = blockIdx.x * 256 + tid;
  const int b = token / L_;

  float lg[E_];
#pragma unroll
  for (int e = 0; e < E_; ++e) lg[e] = br[e];
  const float* xr = x + (size_t)token * D_;
  for (int d = 0; d < D_; ++d) {
    const float xv = xr[d];
#pragma unroll
    for (int e = 0; e < E_; ++e) lg[e] += xv * Wr[d * E_ + e];
  }
#pragma unroll
  for (int t = 0; t < DT_; ++t) {
    const float zv = z_task[b * DT_ + t];
#pragma unroll
    for (int e = 0; e < E_; ++e) lg[e] += zv * Wr[(D_ + t) * E_ + e];
  }
  const float it = 1.0f / tau[b];   // TEMP == 1.0
#pragma unroll
  for (int e = 0; e < E_; ++e) lg[e] *= it;

  // full softmax (importance)
  float mx = lg[0];
#pragma unroll
  for (int e = 1; e < E_; ++e) mx = fmaxf(mx, lg[e]);
  float p[E_], s = 0.0f;
#pragma unroll
  for (int e = 0; e < E_; ++e) { p[e] = expf(lg[e] - mx); s += p[e]; }
  const float is = 1.0f / s;

  // top-2 (strict >, ties -> lowest index like lax.top_k)
  int i1 = 0;
#pragma unroll
  for (int e = 1; e < E_; ++e) if (lg[e] > lg[i1]) i1 = e;
  int i2 = (i1 == 0) ? 1 : 0;
#pragma unroll
  for (int e = 0; e < E_; ++e) if (e != i1 && lg[e] > lg[i2]) i2 = e;

#pragma unroll
  for (int e = 0; e < E_; ++e) atomicAdd(&red[e], p[e] * is);
  atomicAdd(&red[8 + i1], 1.0f);
  atomicAdd(&red[8 + i2], 1.0f);
  __syncthreads();
  if (tid < 16) atomicAdd(&stats[tid], red[tid]);
}

// ---------------- kernel 2: MoE FFN with fp32 WMMA ----------------
// 32 tokens per workgroup, 8 wave32.
//   GEMM1: wave computes h cols [f0+wave*32, +32) for both 16-token M-tiles
//          (B tile loaded once, used by 2 WMMAs -> halved L2 weight traffic);
//          x K-chunks double-buffered via async global->LDS DMA overlapping WMMA.
//   GEMM2: gate is folded into h, so WMMA accumulates y directly across
//          experts and F-chunks in registers (no per-expert C tile).
__global__ __launch_bounds__(256) void tamoe_ffn(
    const float* __restrict__ x, const float* __restrict__ z_task,
    const float* __restrict__ tau, const float* __restrict__ Wr,
    const float* __restrict__ br, const float* __restrict__ W1,
    const float* __restrict__ b1, const float* __restrict__ W2,
    const float* __restrict__ b2, float* __restrict__ y) {
  __shared__ float xs[2 * MW_ * XSC_STRIDE]; // 2 async-staged x K-chunks [32 x 64]
  __shared__ float hs[MW_ * HS_STRIDE];      // gated gelu h chunk [32 x 256]
  __shared__ float lgs[MW_ * E_];            // raw logits
  __shared__ float gs[MW_ * E_];             // dense top-2 gates
  __shared__ int   eflag[E_];                // expert used by any of the 32 tokens?

  const int token0 = blockIdx.x * MW_;
  const int tid    = threadIdx.x;
  const int lane   = tid & 31;
  const int wave   = tid >> 5;               // 0..7
  const int half   = lane >> 4;              // 0 | 1
  const int lm     = lane & 15;

  // ---- in-tile router: 256 threads, one (token m, expert e) logit each ----
  {
    const int m = tid >> 3, e = tid & 7;
    const int b = (token0 + m) / L_;         // all 32 tokens share b (32 | L)
    float acc = br[e];
    const float* xr = x + (size_t)(token0 + m) * D_;
    for (int d = 0; d < D_; ++d) acc += xr[d] * Wr[d * E_ + e];
#pragma unroll
    for (int t = 0; t < DT_; ++t) acc += z_task[b * DT_ + t] * Wr[(D_ + t) * E_ + e];
    lgs[m * E_ + e] = acc / tau[b];          // TEMP == 1.0
  }
  __syncthreads();
  if (tid < MW_) {                           // top-2 + softmax gates per token
    const int m = tid;
    float lg[E_];
#pragma unroll
    for (int e = 0; e < E_; ++e) lg[e] = lgs[m * E_ + e];
    int i1 = 0;
#pragma unroll
    for (int e = 1; e < E_; ++e) if (lg[e] > lg[i1]) i1 = e;
    int i2 = (i1 == 0) ? 1 : 0;
#pragma unroll
    for (int e = 0; e < E_; ++e) if (e != i1 && lg[e] > lg[i2]) i2 = e;
    const float g1 = 1.0f / (1.0f + expf(lg[i2] - lg[i1]));
#pragma unroll
    for (int e = 0; e < E_; ++e) gs[m * E_ + e] = 0.0f;
    gs[m * E_ + i1] = g1;
    gs[m * E_ + i2] = 1.0f - g1;
  }
  __syncthreads();
  if (tid < E_) {
    int f = 0;
    for (int m = 0; m < MW_; ++m) f |= (gs[m * E_ + tid] != 0.0f);
    eflag[tid] = f;
  }
  __syncthreads();

  // y accumulators: [m-tile mt][n-tile t] -> acc[mt*4+t]; wave cols [wave*64,+64)
  v8f acc[8];
#pragma unroll
  for (int t = 0; t < 8; ++t) acc[t] = v8f{};

  for (int e = 0; e < E_; ++e) {
    if (!eflag[e]) continue;                 // uniform across block
    const float* w1 = W1 + (size_t)e * D_ * F_;
    const float* w2 = W2 + (size_t)e * F_ * D_;

    for (int fc = 0; fc < F_ / FC_; ++fc) {
      const int f0  = fc * FC_;
      const int fn0 = f0 + wave * 32;        // this wave's 2 h n-tiles

      // ---- GEMM1: c1[mt][t]; double-buffered async x staging ----
      v8f c1[4];
#pragma unroll
      for (int t = 0; t < 4; ++t) c1[t] = v8f{};

      // stage chunk 0 into buffer 0 (512 b128 DMAs, 2 per thread)
      for (int i = tid; i < MW_ * (KC_ / 4); i += 256) {
        const int m = i >> 4, c4 = i & 15;   // KC_/4 == 16
        async_g2l_b128(&xs[m * XSC_STRIDE + c4 * 4],
                       x + (size_t)(token0 + m) * D_ + c4 * 4);
      }
      wait_async0();
      __syncthreads();

      for (int kc = 0; kc < D_ / KC_; ++kc) {
        const int k0  = kc * KC_;
        const int cur = kc & 1;
        // kick off DMA for next chunk; overlaps with the WMMA loop below
        if (kc + 1 < D_ / KC_) {
          float* nb = xs + (cur ^ 1) * (MW_ * XSC_STRIDE);
          for (int i = tid; i < MW_ * (KC_ / 4); i += 256) {
            const int m = i >> 4, c4 = i & 15;
            async_g2l_b128(&nb[m * XSC_STRIDE + c4 * 4],
                           x + (size_t)(token0 + m) * D_ + (k0 + KC_) + c4 * 4);
          }
          if (lane == 0)                    // hint next W1 k-rows into cache
            __builtin_prefetch(w1 + (size_t)(k0 + KC_) * F_ + fn0, 0, 0);
        }
        const float* xb = xs + cur * (MW_ * XSC_STRIDE);

#pragma unroll 4
        for (int k = 0; k < KC_; k += 4) {
          v2f a0, a1;
          a0.x = xb[lm * XSC_STRIDE + k + 2 * half];
          a0.y = xb[lm * XSC_STRIDE + k + 2 * half + 1];
          a1.x = xb[(16 + lm) * XSC_STRIDE + k + 2 * half];
          a1.y = xb[(16 + lm) * XSC_STRIDE + k + 2 * half + 1];
#pragma unroll
          for (int t = 0; t < 2; ++t) {
            const int n = fn0 + t * 16 + lm;
            v2f bv;                          // B tile shared by both M-tiles
            bv.x = w1[(size_t)(k0 + k + 2 * half) * F_ + n];
            bv.y = w1[(size_t)(k0 + k + 2 * half + 1) * F_ + n];
            c1[t]     = wmma4(a0, bv, c1[t]);
            c1[2 + t] = wmma4(a1, bv, c1[2 + t]);
          }
        }
        wait_async0();                       // next buffer landed
        __syncthreads();                     // and everyone done reading current
      }

      // ---- bias + exact GELU, fold gate, stage chunk to LDS ----
#pragma unroll
      for (int mt = 0; mt < 2; ++mt) {
#pragma unroll
        for (int t = 0; t < 2; ++t) {
          const int n = fn0 + t * 16 + lm;
          const float bb = b1[e * F_ + n];
#pragma unroll
          for (int r = 0; r < 8; ++r) {
            const int m = mt * 16 + r + 8 * half;
            const float v = c1[mt * 2 + t][r] + bb;
            const float g = 0.5f * v * (1.0f + erff(v * 0.70710678118654752f));
            hs[m * HS_STRIDE + (n - f0)] = gs[m * E_ + e] * g;
          }
        }
      }
      __syncthreads();

      // ---- GEMM2 partial: accumulate y in place across experts/chunks ----
#pragma unroll 4
      for (int k = 0; k < FC_; k += 4) {
        v2f a0, a1;
        a0.x = hs[lm * HS_STRIDE + k + 2 * half];
        a0.y = hs[lm * HS_STRIDE + k + 2 * half + 1];
        a1.x = hs[(16 + lm) * HS_STRIDE + k + 2 * half];
        a1.y = hs[(16 + lm) * HS_STRIDE + k + 2 * half + 1];
#pragma unroll
        for (int t = 0; t < 4; ++t) {
          const int n = wave * 64 + t * 16 + lm;
          v2f bv;                            // B tile shared by both M-tiles
          bv.x = w2[(size_t)(f0 + k + 2 * half) * D_ + n];
          bv.y = w2[(size_t)(f0 + k + 2 * half + 1) * D_ + n];
          acc[t]     = wmma4(a0, bv, acc[t]);
          acc[4 + t] = wmma4(a1, bv, acc[4 + t]);
        }
      }
      __syncthreads();                       // hs rewritten next chunk
    }

    // gate-scaled expert bias: y += g[m] * b2[e]
#pragma unroll
    for (int mt = 0; mt < 2; ++mt) {
#pragma unroll
      for (int t = 0; t < 4; ++t) {
        const int n = wave * 64 + t * 16 + lm;
        const float bb = b2[e * D_ + n];
#pragma unroll
        for (int r = 0; r < 8; ++r) {
          const int m = mt * 16 + r + 8 * half;
          acc[mt * 4 + t][r] += gs[m * E_ + e] * bb;
        }
      }
    }
  }

  // ---- write y (each (m,n) owned by exactly one lane/vgpr) ----
#pragma unroll
  for (int mt = 0; mt < 2; ++mt) {
#pragma unroll
    for (int t = 0; t < 4; ++t) {
      const int n = wave * 64 + t * 16 + lm;
#pragma unroll
      for (int r = 0; r < 8; ++r) {
        const int m = mt * 16 + r + 8 * half;
        y[(size_t)(token0 + m) * D_ + n] = acc[mt * 4 + t][r];
      }
    }
  }
}

// ---------------- kernel 3: aux loss ----------------
__global__ void tamoe_aux(const float* __restrict__ stats, float* __restrict__ out) {
  if (threadIdx.x == 0 && blockIdx.x == 0) {
    const float inv = 1.0f / (float)NT_;
    float s = 0.0f;
#pragma unroll
    for (int e = 0; e < E_; ++e)
      s += (stats[8 + e] * inv) * (stats[e] * inv);
    out[(size_t)NT_ * D_] = (float)E_ * s;
  }
}

// ---------------- launcher ----------------
extern "C" void kernel_launch(void* const* d_in, const int* in_sizes, int n_in,
                              void* d_out, int out_size, void* d_ws, size_t ws_size,
                              hipStream_t stream) {
  const float* x      = (const float*)d_in[0];
  const float* z_task = (const float*)d_in[1];
  const float* tau    = (const float*)d_in[2];
  const float* Wr     = (const float*)d_in[3];
  const float* br     = (const float*)d_in[4];
  const float* W1     = (const float*)d_in[5];
  const float* b1     = (const float*)d_in[6];
  const float* W2     = (const float*)d_in[7];
  const float* b2     = (const float*)d_in[8];
  float* out   = (float*)d_out;
  float* stats = (float*)d_ws;   // 16 floats of aux accumulators

  tamoe_init<<<1, 32, 0, stream>>>(stats);
  tamoe_router_stats<<<NT_ / 256, 256, 0, stream>>>(x, z_task, tau, Wr, br, stats);
  tamoe_ffn<<<NT_ / MW_, 256, 0, stream>>>(x, z_task, tau, Wr, br, W1, b1, W2, b2, out);
  tamoe_aux<<<1, 1, 0, stream>>>(stats, out);
}